// DescriptorBuilder_11338713661485
// MI455X (gfx1250) — compile-verified
//
#include <hip/hip_runtime.h>
#include <hip/hip_bf16.h>
#include <math.h>

// MI455X / gfx1250, wave32. One 256-thread block (8 waves) per center atom.
// Angular 3-body Gram matrices via V_WMMA_F32_16X16X4_F32 (rank-3 SYRK, K=4 padded).
// Symmetric (upper-triangular) tiling: 78 tiles, off-diagonal weighted x2.

typedef float v2f __attribute__((ext_vector_type(2)));
typedef float v8f __attribute__((ext_vector_type(8)));

#define NATOMS   192
#define NTILES   12          // 192 / 16
#define NTRI     78          // NTILES*(NTILES+1)/2 upper-triangular tiles
#define RC_F     6.0f
#define KRAD     9           // radial powers k = 0..8
#define PI_F     3.14159265358979323846f

__global__ __launch_bounds__(256)
void descr_kernel(const float* __restrict__ R,
                  const int*   __restrict__ Z,   // unused (kept for extensibility)
                  const float* __restrict__ box,
                  float*       __restrict__ out)
{
    __shared__ float Dld[NATOMS][4];   // dr_ij padded with 0 in component 3
    __shared__ float rl [NATOMS];      // r_ij (with reference's +1e-16 per component)
    __shared__ float fl [NATOMS];      // fc(r_ij) * (j != i)
    __shared__ float qr [KRAD];        // radial accumulators
    __shared__ float qa [9];           // angular accumulators (n*3 + l)

    const int i    = blockIdx.x;
    const int tid  = threadIdx.x;
    const int lane = tid & 31;
    const int wave = tid >> 5;

    if (tid < KRAD) qr[tid] = 0.0f;
    if (tid < 9)    qa[tid] = 0.0f;

    // ---- 3x3 box inverse (redundant per-thread; 9 scalar loads, trivial) ----
    const float b00 = box[0], b01 = box[1], b02 = box[2];
    const float b10 = box[3], b11 = box[4], b12 = box[5];
    const float b20 = box[6], b21 = box[7], b22 = box[8];
    const float det = b00 * (b11 * b22 - b12 * b21)
                    - b01 * (b10 * b22 - b12 * b20)
                    + b02 * (b10 * b21 - b11 * b20);
    const float id  = 1.0f / det;
    const float i00 =  (b11 * b22 - b12 * b21) * id;
    const float i01 = -(b01 * b22 - b02 * b21) * id;
    const float i02 =  (b01 * b12 - b02 * b11) * id;
    const float i10 = -(b10 * b22 - b12 * b20) * id;
    const float i11 =  (b00 * b22 - b02 * b20) * id;
    const float i12 = -(b00 * b12 - b02 * b10) * id;
    const float i20 =  (b10 * b21 - b11 * b20) * id;
    const float i21 = -(b00 * b21 - b01 * b20) * id;
    const float i22 =  (b00 * b11 - b01 * b10) * id;

    // fractional coords of center i
    const float Rix = R[3 * i + 0], Riy = R[3 * i + 1], Riz = R[3 * i + 2];
    const float six = i00 * Rix + i01 * Riy + i02 * Riz;
    const float siy = i10 * Rix + i11 * Riy + i12 * Riz;
    const float siz = i20 * Rix + i21 * Riy + i22 * Riz;

    // ---- Phase 1: per-neighbor MIC displacement, r, fc ; radial features ----
    float fcm_j = 0.0f, r_j = 0.0f;
    if (tid < NATOMS) {
        const int j = tid;
        const float Rjx = R[3 * j + 0], Rjy = R[3 * j + 1], Rjz = R[3 * j + 2];
        float dsx = six - (i00 * Rjx + i01 * Rjy + i02 * Rjz);
        float dsy = siy - (i10 * Rjx + i11 * Rjy + i12 * Rjz);
        float dsz = siz - (i20 * Rjx + i21 * Rjy + i22 * Rjz);
        dsx -= rintf(dsx);                 // round-half-even == jnp.round
        dsy -= rintf(dsy);
        dsz -= rintf(dsz);
        const float dx = b00 * dsx + b01 * dsy + b02 * dsz;
        const float dy = b10 * dsx + b11 * dsy + b12 * dsz;
        const float dz = b20 * dsx + b21 * dsy + b22 * dsz;

        // reference: rij = norm(dr + 1e-16)
        const float ex = dx + 1e-16f, ey = dy + 1e-16f, ez = dz + 1e-16f;
        r_j = sqrtf(ex * ex + ey * ey + ez * ez);

        float x = r_j * (1.0f / RC_F);
        x = fminf(1.0f, fmaxf(0.0f, x));
        const float fc = 0.5f * (__cosf(PI_F * x) + 1.0f);   // v_cos_f32
        fcm_j = (j == i) ? 0.0f : fc;

        Dld[j][0] = dx; Dld[j][1] = dy; Dld[j][2] = dz; Dld[j][3] = 0.0f;
        rl[j] = r_j;
        fl[j] = fcm_j;
    }
    __syncthreads();   // qr/qa initialized; Dld/rl/fl visible to all waves

    if (tid < NATOMS) {
        float pk = 1.0f;                   // r^k, k = 0..8
        #pragma unroll
        for (int k = 0; k < KRAD; ++k) {
            atomicAdd(&qr[k], pk * fcm_j); // ds_add_f32
            pk *= r_j;
        }
    }

    // ---- Phase 2: 3-body term over upper-triangular Gram tiles (tj <= tk) ----
    float acc[9];
    #pragma unroll
    for (int c = 0; c < 9; ++c) acc[c] = 0.0f;

    const int m16   = lane & 15;
    const int k0    = (lane < 16) ? 0 : 2;   // ISA 16x4 f32 A/B fragment layout
    const int mbase = (lane < 16) ? 0 : 8;   // ISA 16x16 f32 C/D layout

    int tj = 0, rowStart = 0, rowEnd = NTILES;     // row tj covers [rowStart,rowEnd)
    for (int t = wave; t < NTRI; t += 8) {
        while (t >= rowEnd) { ++tj; rowStart = rowEnd; rowEnd += NTILES - tj; }
        const int tk = tj + (t - rowStart);

        v2f a, b;
        const int rA = tj * 16 + m16;
        const int rB = tk * 16 + m16;
        a.x = Dld[rA][k0];  a.y = Dld[rA][k0 + 1];
        b.x = Dld[rB][k0];  b.y = Dld[rB][k0 + 1];

        v8f g = {};
        // D = A(16x4) * B(4x16) + 0 : G[m][n] = dr_j(m) . dr_k(n)
        g = __builtin_amdgcn_wmma_f32_16x16x4_f32(
                /*neg_a=*/false, a, /*neg_b=*/false, b,
                /*c_mod=*/(short)0, g, /*reuse_a=*/false, /*reuse_b=*/false);

        const float scale = (tj == tk) ? 1.0f : 2.0f;   // symmetry weight
        const int   gk  = tk * 16 + m16;                // column -> neighbor k
        const float rk  = rl[gk];
        const float fks = fl[gk] * scale;

        #pragma unroll
        for (int v = 0; v < 8; ++v) {
            const int   gj = tj * 16 + mbase + v;       // row -> neighbor j
            const float f  = fl[gj] * fks;              // fcm_ij * fcm_ik * weight
            const float rp = rl[gj] * rk;               // r_ij * r_ik
            const float inv = __builtin_amdgcn_rcpf(rp + 1e-8f);   // v_rcp_f32
            float cth = g[v] * inv;
            cth = fminf(1.0f, fmaxf(-1.0f, cth));
            const float p2  = 0.5f * (3.0f * cth * cth - 1.0f);
            const float rp2 = rp * rp;
            acc[0] += f;                 // n=0, P0
            acc[1] += f * cth;           // n=0, P1
            acc[2] += f * p2;            // n=0, P2
            acc[3] += f * rp;            // n=1, P0
            acc[4] += f * rp * cth;      // n=1, P1
            acc[5] += f * rp * p2;       // n=1, P2
            acc[6] += f * rp2;           // n=2, P0
            acc[7] += f * rp2 * cth;     // n=2, P1
            acc[8] += f * rp2 * p2;      // n=2, P2
        }
    }

    #pragma unroll
    for (int c = 0; c < 9; ++c) atomicAdd(&qa[c], acc[c]);
    __syncthreads();

    // ---- Phase 3: write [q_r (9) | q_ang (9)] for center i ----
    if (tid < KRAD)          out[i * 18 + tid] = qr[tid];
    else if (tid < 18)       out[i * 18 + tid] = qa[tid - KRAD];
}

extern "C" void kernel_launch(void* const* d_in, const int* in_sizes, int n_in,
                              void* d_out, int out_size, void* d_ws, size_t ws_size,
                              hipStream_t stream) {
    const float* R   = (const float*)d_in[0];   // [192,3]
    const int*   Z   = (const int*)  d_in[1];   // [192] (unused)
    const float* box = (const float*)d_in[2];   // [3,3]
    float*       out = (float*)d_out;           // [192,18]
    (void)in_sizes; (void)n_in; (void)d_ws; (void)ws_size; (void)out_size;

    descr_kernel<<<NATOMS, 256, 0, stream>>>(R, Z, box, out);
}